// MTGNN_29764123361449
// MI455X (gfx1250) — compile-verified
//
#include <hip/hip_runtime.h>
#include <hip/hip_bf16.h>
#include <math.h>

// ---------------------------------------------------------------------------
// MTGNN layer forward for MI455X (gfx1250, wave32).
// Graph diffusion / skip conv / channel-mix run on V_WMMA_F32_16X16X4_F32
// (full fp32 on the matrix pipe). Intermediate buffers pad time TOUT=162 to
// LP=176 (11 full 16-tiles) so WMMA inner loops carry no predication.
// The prop kernel stages its adjacency strip into LDS with gfx1250
// GLOBAL_LOAD_ASYNC_TO_LDS_B128 (ASYNCcnt-tracked async copies).
// ---------------------------------------------------------------------------

typedef __attribute__((ext_vector_type(2))) float v2f;
typedef __attribute__((ext_vector_type(8))) float v8f;

#define BB   16
#define RR   32
#define CC   32
#define SCH  64
#define NN   500
#define TT   168
#define TOUT 162
#define LP   176                 // padded time stride (11 * 16)
#define POS  (NN * TOUT)         // 81000 valid positions per (b, channel)
#define POSP (NN * LP)           // 88000 padded positions
#define PB_PER_B (CC * POS)      // 2,592,000 valid elements per batch sample
#define KSKIP (CC * TOUT)        // 5184
#define ALPHA 0.05f

static __device__ __forceinline__ v8f wmma_f32_k4(v2f a, v2f b, v8f c) {
  return __builtin_amdgcn_wmma_f32_16x16x4_f32(false, a, false, b, (short)0, c,
                                               false, false);
}

// ---------------------------------------------------------------------------
// 1) Adjacency normalization: a1 = rownorm(adj + I), a2 = rownorm(adj^T + I)
// ---------------------------------------------------------------------------
__global__ __launch_bounds__(256) void adjnorm_kernel(
    const float* __restrict__ adj, float* __restrict__ a1,
    float* __restrict__ a2) {
  const int v = blockIdx.x;
  __shared__ float red[256];
  __shared__ float rs1s, rs2s;

  float s = 0.f;
  for (int w = threadIdx.x; w < NN; w += 256) s += adj[v * NN + w];
  red[threadIdx.x] = s;
  __syncthreads();
  for (int st = 128; st > 0; st >>= 1) {
    if (threadIdx.x < st) red[threadIdx.x] += red[threadIdx.x + st];
    __syncthreads();
  }
  if (threadIdx.x == 0) rs1s = red[0] + 1.0f;
  __syncthreads();

  float s2 = 0.f;
  for (int w = threadIdx.x; w < NN; w += 256) s2 += adj[w * NN + v];
  red[threadIdx.x] = s2;
  __syncthreads();
  for (int st = 128; st > 0; st >>= 1) {
    if (threadIdx.x < st) red[threadIdx.x] += red[threadIdx.x + st];
    __syncthreads();
  }
  if (threadIdx.x == 0) rs2s = red[0] + 1.0f;
  __syncthreads();

  const float rs1 = rs1s, rs2 = rs2s;
  for (int w = threadIdx.x; w < NN; w += 256) {
    const float d = (w == v) ? 1.0f : 0.0f;
    a1[v * NN + w] = (adj[v * NN + w] + d) / rs1;
    a2[v * NN + w] = (adj[w * NN + v] + d) / rs2;
  }
}

// ---------------------------------------------------------------------------
// 2) Dilated inception + gating -> h[b,c,n,l] (padded stride LP, pad zeroed)
// ---------------------------------------------------------------------------
__global__ __launch_bounds__(256) void inception_kernel(
    const float* __restrict__ x,
    const float* __restrict__ fw2, const float* __restrict__ fb2,
    const float* __restrict__ gw2, const float* __restrict__ gb2,
    const float* __restrict__ fw3, const float* __restrict__ fb3,
    const float* __restrict__ gw3, const float* __restrict__ gb3,
    const float* __restrict__ fw6, const float* __restrict__ fb6,
    const float* __restrict__ gw6, const float* __restrict__ gb6,
    const float* __restrict__ fw7, const float* __restrict__ fb7,
    const float* __restrict__ gw7, const float* __restrict__ gb7,
    float* __restrict__ H) {
  const int b = blockIdx.x;
  const int n = blockIdx.y;
  const int tid = threadIdx.x;

  __shared__ float xs[RR * TT];
  __shared__ float fwS[4608];
  __shared__ float gwS[4608];
  __shared__ float fbS[32];
  __shared__ float gbS[32];

  for (int i = tid; i < RR * TT; i += 256) {
    const int r = i / TT, t = i % TT;
    xs[i] = x[(((size_t)b * RR + r) * NN + n) * TT + t];
  }
  for (int i = tid; i < 512; i += 256)  { fwS[i]        = fw2[i]; gwS[i]        = gw2[i]; }
  for (int i = tid; i < 768; i += 256)  { fwS[512 + i]  = fw3[i]; gwS[512 + i]  = gw3[i]; }
  for (int i = tid; i < 1536; i += 256) { fwS[1280 + i] = fw6[i]; gwS[1280 + i] = gw6[i]; }
  for (int i = tid; i < 1792; i += 256) { fwS[2816 + i] = fw7[i]; gwS[2816 + i] = gw7[i]; }
  if (tid < 8) {
    fbS[tid] = fb2[tid];      gbS[tid] = gb2[tid];
    fbS[8 + tid] = fb3[tid];  gbS[8 + tid] = gb3[tid];
    fbS[16 + tid] = fb6[tid]; gbS[16 + tid] = gb6[tid];
    fbS[24 + tid] = fb7[tid]; gbS[24 + tid] = gb7[tid];
  }
  __syncthreads();

  const int ksz[4]  = {2, 3, 6, 7};
  const int offs[4] = {0, 512, 1280, 2816};

  float* hbn = H + (size_t)b * CC * NN * LP + (size_t)n * LP;

  for (int idx = tid; idx < CC * TOUT; idx += 256) {
    const int c = idx / TOUT, t = idx % TOUT;
    const int br = c >> 3, o = c & 7;
    const int k = ksz[br];
    const int woff = offs[br] + o * 32 * k;
    float fs = fbS[br * 8 + o];
    float gs = gbS[br * 8 + o];
    const int tbase = t + 7 - k;  // window always ends at t+6
    for (int r = 0; r < RR; ++r) {
      const float* xr = &xs[r * TT + tbase];
      const float* wf = &fwS[woff + r * k];
      const float* wg = &gwS[woff + r * k];
      for (int j = 0; j < k; ++j) {
        const float xv = xr[j];
        fs += xv * wf[j];
        gs += xv * wg[j];
      }
    }
    hbn[(size_t)c * NN * LP + t] = tanhf(fs) * (1.0f / (1.0f + expf(-gs)));
  }
  // zero the pad columns (keeps padded WMMA lanes finite & deterministic)
  for (int i = tid; i < CC * (LP - TOUT); i += 256) {
    const int c = i / (LP - TOUT), l = TOUT + i % (LP - TOUT);
    hbn[(size_t)c * NN * LP + l] = 0.0f;
  }
}

// ---------------------------------------------------------------------------
// 3) Skip conv via WMMA: S[b,o,n] = sum_{c,t} h*skip_w + sb.
//    c-outer / t-inner: linear addresses, b64 loads, one masked remainder
//    WMMA per channel for t=160,161. Out-of-range n lanes read clamped
//    addresses; their garbage D-columns are never stored (columns of a
//    matmul are independent).
// ---------------------------------------------------------------------------
__global__ __launch_bounds__(128) void skip_kernel(
    const float* __restrict__ H, const float* __restrict__ SW,
    const float* __restrict__ SB, float* __restrict__ S) {
  const int b = blockIdx.x;
  const int n0 = blockIdx.y * 16;
  const int wave = threadIdx.x >> 5;  // o tile 0..3
  const int lane = threadIdx.x & 31;
  const int o0 = wave * 16;
  const int m = lane & 15;
  const int kb = (lane >> 4) << 1;  // 0 or 2

  const int n = n0 + m;
  const int n_safe = (n < NN) ? n : (NN - 1);

  const float* swrow = SW + (size_t)(o0 + m) * KSKIP + kb;
  const float* hbn = H + (size_t)b * CC * NN * LP + (size_t)n_safe * LP + kb;

  v8f acc = {};
  for (int c = 0; c < CC; ++c) {
    const float* swc = swrow + c * TOUT;
    const float* hc = hbn + (size_t)c * NN * LP;
#pragma unroll 4
    for (int t = 0; t < 160; t += 4) {
      const v2f a = *(const v2f*)(swc + t);
      const v2f bv = *(const v2f*)(hc + t);
      acc = wmma_f32_k4(a, bv, acc);
    }
    // remainder t = 160,161: only the kb==0 half-wave carries data
    v2f a = {0.f, 0.f}, bv = {0.f, 0.f};
    if (kb == 0) {
      a = *(const v2f*)(swc + 160);
      bv = *(const v2f*)(hc + 160);
    }
    acc = wmma_f32_k4(a, bv, acc);
  }

  const int nc = n0 + (lane & 15);
  const int mbase = (lane >> 4) * 8;
  if (nc < NN) {
#pragma unroll
    for (int i = 0; i < 8; ++i) {
      const int o = o0 + mbase + i;
      S[((size_t)b * SCH + o) * NN + nc] = acc[i] + SB[o];
    }
  }
}

// ---------------------------------------------------------------------------
// 4) Graph propagation via WMMA with fused mix-hop blend (padded layout):
//    Out = ALPHA*Hbase + (1-ALPHA)*(A @ Hin), per (b,c) slice [500 x LP].
//    Block: one (b*c, vtile). The 16x500 A strip is staged into LDS with
//    gfx1250 async global->LDS B128 copies (ASYNCcnt), then reused by all
//    8 waves x 11 l-tiles x 125 K-steps.
// ---------------------------------------------------------------------------
__global__ __launch_bounds__(256) void prop_kernel(
    const float* __restrict__ A, const float* __restrict__ Hin,
    const float* __restrict__ Hbase, float* __restrict__ Out) {
  __shared__ float As[16 * NN];  // 32 KB
  const int bc = blockIdx.x;
  const int v0 = blockIdx.y * 16;

  // Async stage A[v0:v0+16, 0:500] -> LDS. Rows are 16B-aligned (500%4==0).
  for (int q = threadIdx.x; q < 16 * (NN / 4); q += 256) {
    const int r = q / (NN / 4);
    const int w4 = (q - r * (NN / 4)) * 4;
    const int row = v0 + r;
    float* lp = &As[r * NN + w4];
    if (row < NN) {
      const unsigned lds_off = (unsigned)(size_t)lp;
      const unsigned long long gaddr =
          (unsigned long long)(size_t)(A + (size_t)row * NN + w4);
      asm volatile("global_load_async_to_lds_b128 %0, %1, off"
                   :
                   : "v"(lds_off), "v"(gaddr)
                   : "memory");
    } else {
      lp[0] = 0.f; lp[1] = 0.f; lp[2] = 0.f; lp[3] = 0.f;
    }
  }
  asm volatile("s_wait_asynccnt 0x0" ::: "memory");
  __syncthreads();

  const int wave = threadIdx.x >> 5;
  const int lane = threadIdx.x & 31;
  const int m = lane & 15;
  const int kb = (lane >> 4) << 1;
  const int mbase = (lane >> 4) * 8;

  const float* hin_bc = Hin + (size_t)bc * NN * LP;
  const float* hb_bc = Hbase + (size_t)bc * NN * LP;
  float* out_bc = Out + (size_t)bc * NN * LP;
  const float* ap = &As[m * NN + kb];  // m*NN even, kb even -> 8B aligned

  for (int lt = wave; lt < 11; lt += 8) {
    const int l0 = lt * 16;
    const float* hp = hin_bc + (size_t)kb * LP + l0 + m;
    v8f acc = {};
#pragma unroll 5
    for (int k = 0; k < NN; k += 4) {
      const v2f a = *(const v2f*)(ap + k);
      v2f bv;
      bv.x = hp[0];
      bv.y = hp[LP];
      hp += 4 * LP;
      acc = wmma_f32_k4(a, bv, acc);
    }
    const int col = l0 + m;
#pragma unroll
    for (int i = 0; i < 8; ++i) {
      const int row = v0 + mbase + i;
      if (row < NN) {
        const size_t idx = (size_t)row * LP + col;
        out_bc[idx] = ALPHA * hb_bc[idx] + (1.0f - ALPHA) * acc[i];
      }
    }
  }
}

// ---------------------------------------------------------------------------
// 5) Channel-mix projection via WMMA (K=32, fully unrolled -> 8 WMMAs):
//    init: Y = (W1+W2)[:,coloff:+32] @ P + b1+b2 + residual-crop(x)
//    acc : Y += W1[:,coloff:+32] @ P
// ---------------------------------------------------------------------------
__global__ __launch_bounds__(256) void chanmix_kernel(
    float* __restrict__ Y, const float* __restrict__ P,
    const float* __restrict__ W1, const float* __restrict__ W2, int coloff,
    const float* __restrict__ B1, const float* __restrict__ B2,
    const float* __restrict__ Xc, int init) {
  const int b = blockIdx.x;
  const int o0 = blockIdx.y * 16;
  const int wave = threadIdx.x >> 5;
  const int lane = threadIdx.x & 31;
  const int pt = blockIdx.z * 8 + wave;
  const int p0 = pt * 16;
  if (p0 >= POSP) return;

  const int m = lane & 15;
  const int kb = (lane >> 4) << 1;
  const float* pp = P + (size_t)b * CC * POSP + (size_t)kb * POSP + p0 + m;
  const int wbase = (o0 + m) * 96 + coloff + kb;  // even -> 8B aligned

  v8f acc = {};
#pragma unroll
  for (int k = 0; k < CC; k += 4) {
    v2f a = *(const v2f*)(W1 + wbase + k);
    if (W2) {
      const v2f a2 = *(const v2f*)(W2 + wbase + k);
      a.x += a2.x;
      a.y += a2.y;
    }
    v2f bv;
    bv.x = pp[0];
    bv.y = pp[POSP];
    pp += 4 * (size_t)POSP;
    acc = wmma_f32_k4(a, bv, acc);
  }

  const int pc = p0 + (lane & 15);
  const int mbase = (lane >> 4) * 8;
  const int v = pc / LP;
  const int l = pc - v * LP;
#pragma unroll
  for (int i = 0; i < 8; ++i) {
    const int o = o0 + mbase + i;
    const size_t yi = ((size_t)b * CC + o) * POSP + pc;
    float r = acc[i];
    if (init) {
      r += B1[o] + B2[o];
      if (l < TOUT)
        r += Xc[(((size_t)b * RR + o) * NN + v) * TT + 6 + l];
      Y[yi] = r;
    } else {
      Y[yi] += r;
    }
  }
}

// ---------------------------------------------------------------------------
// 6) LayerNorm: deterministic two-stage reduction + apply with idx gather.
//    y is padded [b][c][v][LP]; only l < TOUT participates.
// ---------------------------------------------------------------------------
#define PBLK 160
#define ROWS_BLK 100  // (CC*NN)/PBLK = 16000/160 rows per block

__global__ __launch_bounds__(256) void ln_partial_kernel(
    const float* __restrict__ Y, float* __restrict__ partials) {
  const int b = blockIdx.x;
  const int blk = blockIdx.y;
  const float* yb =
      Y + (size_t)b * CC * POSP + (size_t)blk * ROWS_BLK * LP;
  float s = 0.f, ss = 0.f;
  for (int rr = 0; rr < ROWS_BLK; ++rr) {
    const float* rp = yb + (size_t)rr * LP;
    for (int l = threadIdx.x; l < TOUT; l += 256) {
      const float v = rp[l];
      s += v;
      ss += v * v;
    }
  }
  __shared__ float rs[256], rss[256];
  rs[threadIdx.x] = s;
  rss[threadIdx.x] = ss;
  __syncthreads();
  for (int st = 128; st > 0; st >>= 1) {
    if (threadIdx.x < st) {
      rs[threadIdx.x] += rs[threadIdx.x + st];
      rss[threadIdx.x] += rss[threadIdx.x + st];
    }
    __syncthreads();
  }
  if (threadIdx.x == 0) {
    partials[(b * PBLK + blk) * 2 + 0] = rs[0];
    partials[(b * PBLK + blk) * 2 + 1] = rss[0];
  }
}

__global__ __launch_bounds__(256) void ln_final_kernel(
    const float* __restrict__ partials, float* __restrict__ stats) {
  const int b = blockIdx.x;
  float s = 0.f, ss = 0.f;
  for (int i = threadIdx.x; i < PBLK; i += 256) {
    s += partials[(b * PBLK + i) * 2 + 0];
    ss += partials[(b * PBLK + i) * 2 + 1];
  }
  __shared__ float rs[256], rss[256];
  rs[threadIdx.x] = s;
  rss[threadIdx.x] = ss;
  __syncthreads();
  for (int st = 128; st > 0; st >>= 1) {
    if (threadIdx.x < st) {
      rs[threadIdx.x] += rs[threadIdx.x + st];
      rss[threadIdx.x] += rss[threadIdx.x + st];
    }
    __syncthreads();
  }
  if (threadIdx.x == 0) {
    const float inv = 1.0f / (float)PB_PER_B;
    const float mu = rs[0] * inv;
    const float var = rss[0] * inv - mu * mu;
    stats[b * 2 + 0] = mu;
    stats[b * 2 + 1] = rsqrtf(var + 1e-5f);
  }
}

__global__ __launch_bounds__(256) void ln_apply_kernel(
    const float* __restrict__ Y, const float* __restrict__ stats,
    const float* __restrict__ lnw, const float* __restrict__ lnb,
    const int* __restrict__ idx, float* __restrict__ Out) {
  const int total = BB * PB_PER_B;
  for (int i = blockIdx.x * blockDim.x + threadIdx.x; i < total;
       i += gridDim.x * blockDim.x) {
    const int b = i / PB_PER_B;
    const int rem = i % PB_PER_B;
    const int c = rem / POS;
    const int rem2 = rem % POS;
    const int v = rem2 / TOUT;
    const int l = rem2 % TOUT;
    const float mu = stats[b * 2 + 0];
    const float rstd = stats[b * 2 + 1];
    const size_t yi = (((size_t)b * CC + c) * NN + v) * LP + l;
    const int gi = (c * NN + idx[v]) * TOUT + l;
    Out[i] = (Y[yi] - mu) * rstd * lnw[gi] + lnb[gi];
  }
}

// ---------------------------------------------------------------------------
// Host-side orchestration
// ---------------------------------------------------------------------------
extern "C" void kernel_launch(void* const* d_in, const int* in_sizes, int n_in,
                              void* d_out, int out_size, void* d_ws,
                              size_t ws_size, hipStream_t stream) {
  const float* x      = (const float*)d_in[0];
  const float* adj    = (const float*)d_in[1];
  const float* fw2    = (const float*)d_in[2];
  const float* fb2    = (const float*)d_in[3];
  const float* gw2    = (const float*)d_in[4];
  const float* gb2    = (const float*)d_in[5];
  const float* fw3    = (const float*)d_in[6];
  const float* fb3    = (const float*)d_in[7];
  const float* gw3    = (const float*)d_in[8];
  const float* gb3    = (const float*)d_in[9];
  const float* fw6    = (const float*)d_in[10];
  const float* fb6    = (const float*)d_in[11];
  const float* gw6    = (const float*)d_in[12];
  const float* gb6    = (const float*)d_in[13];
  const float* fw7    = (const float*)d_in[14];
  const float* fb7    = (const float*)d_in[15];
  const float* gw7    = (const float*)d_in[16];
  const float* gb7    = (const float*)d_in[17];
  const float* skip_w = (const float*)d_in[18];
  const float* skip_b = (const float*)d_in[19];
  const float* gc1_w  = (const float*)d_in[20];
  const float* gc1_b  = (const float*)d_in[21];
  const float* gc2_w  = (const float*)d_in[22];
  const float* gc2_b  = (const float*)d_in[23];
  const float* ln_w   = (const float*)d_in[24];
  const float* ln_b   = (const float*)d_in[25];
  const int*   idx    = (const int*)d_in[26];

  const size_t HSZP = (size_t)BB * CC * POSP;      // padded buffer (floats)
  const size_t HSZ  = (size_t)BB * CC * NN * TOUT; // valid output elements
  float* ws = (float*)d_ws;
  float* a1 = ws;
  float* a2 = a1 + (size_t)NN * NN;
  float* h  = a2 + (size_t)NN * NN;
  float* p1 = h + HSZP;
  float* p2 = p1 + HSZP;
  float* y  = p2 + HSZP;
  float* partials = y + HSZP;
  float* stats = partials + 16 * PBLK * 2;

  float* out_yn = (float*)d_out;  // [B, C, N, TOUT]
  float* out_s  = out_yn + HSZ;   // [B, SCH, N, 1]

  adjnorm_kernel<<<NN, 256, 0, stream>>>(adj, a1, a2);

  inception_kernel<<<dim3(BB, NN), 256, 0, stream>>>(
      x, fw2, fb2, gw2, gb2, fw3, fb3, gw3, gb3, fw6, fb6, gw6, gb6, fw7, fb7,
      gw7, gb7, h);

  skip_kernel<<<dim3(BB, 32), 128, 0, stream>>>(h, skip_w, skip_b, out_s);

  const dim3 cmGrid(BB, 2, (POSP / 16 + 7) / 8);  // 5500 pos-tiles / 8 waves
  const dim3 prGrid(BB * CC, 32, 1);              // 32 v-tiles, lt loop inside

  chanmix_kernel<<<cmGrid, 256, 0, stream>>>(y, h, gc1_w, gc2_w, 0, gc1_b,
                                             gc2_b, x, 1);
  prop_kernel<<<prGrid, 256, 0, stream>>>(a1, h, h, p1);
  chanmix_kernel<<<cmGrid, 256, 0, stream>>>(y, p1, gc1_w, nullptr, 32,
                                             nullptr, nullptr, nullptr, 0);
  prop_kernel<<<prGrid, 256, 0, stream>>>(a1, p1, h, p2);
  chanmix_kernel<<<cmGrid, 256, 0, stream>>>(y, p2, gc1_w, nullptr, 64,
                                             nullptr, nullptr, nullptr, 0);
  prop_kernel<<<prGrid, 256, 0, stream>>>(a2, h, h, p1);
  chanmix_kernel<<<cmGrid, 256, 0, stream>>>(y, p1, gc2_w, nullptr, 32,
                                             nullptr, nullptr, nullptr, 0);
  prop_kernel<<<prGrid, 256, 0, stream>>>(a2, p1, h, p2);
  chanmix_kernel<<<cmGrid, 256, 0, stream>>>(y, p2, gc2_w, nullptr, 64,
                                             nullptr, nullptr, nullptr, 0);

  ln_partial_kernel<<<dim3(BB, PBLK), 256, 0, stream>>>(y, partials);
  ln_final_kernel<<<BB, 256, 0, stream>>>(partials, stats);
  ln_apply_kernel<<<20000, 256, 0, stream>>>(y, stats, ln_w, ln_b, idx,
                                             out_yn);
}